// CFRPWaveSimulatorRicker_10668698763490
// MI455X (gfx1250) — compile-verified
//
#include <hip/hip_runtime.h>
#include <stdint.h>

// ---------------- problem constants ----------------
#define NXg 512
#define NYg 512
#define NTg 240
#define STRIDEg 4
#define FRAMES (NTg / STRIDEg)          // 60
#define FRAME_ELEMS (NXg * NYg)

// padded field layout (float2 = {ux, uy} interleaved):
// interior (i,j) -> padded (i+1, j+1)
#define LDA  520                         // padded row stride in float2 elements
#define PADH 516                         // padded rows allocated (>=514)

#define TILE  32
#define HTILE 34                         // tile + 1-halo each side

typedef unsigned int v4u __attribute__((ext_vector_type(4)));
typedef int          v4i __attribute__((ext_vector_type(4)));
typedef int          v8i __attribute__((ext_vector_type(8)));

// ---------------- TDM tile load: global (strided 2D, 8B elems) -> LDS ----------------
#if defined(__has_builtin)
#if __has_builtin(__builtin_amdgcn_tensor_load_to_lds)
#define HAVE_TDM 1
#endif
#endif

__device__ __forceinline__ void tdm_load_tile8(const void* gsrc, void* ldst) {
#ifdef HAVE_TDM
  uint64_t ga = (uint64_t)(uintptr_t)gsrc;
  uint32_t la = (uint32_t)(uintptr_t)ldst;      // low 32 bits of generic LDS ptr = LDS byte addr

  v4u g0;
  g0.x = 1u;                                     // count=1, user descriptor
  g0.y = la;                                     // lds_addr
  g0.z = (uint32_t)ga;                           // global_addr[31:0]
  g0.w = (uint32_t)((ga >> 32) & 0x01FFFFFFu) | (2u << 30);  // global_addr[56:32] | type=2

  v8i g1;
  g1[0] = (int)(3u << 16);                                   // workgroup_mask=0, data_size=3 (8B)
  g1[1] = (int)(((unsigned)LDA & 0xFFFFu) << 16);            // tensor_dim0 lo16 (bits 63:48)
  g1[2] = (int)(((unsigned)LDA >> 16) |
                (((unsigned)PADH & 0xFFFFu) << 16));         // dim0 hi16 | tensor_dim1 lo16
  g1[3] = (int)(((unsigned)PADH >> 16) |
                ((unsigned)HTILE << 16));                    // dim1 hi16 | tile_dim0 = 34
  g1[4] = (int)((unsigned)HTILE);                            // tile_dim1 = 34, tile_dim2 = 0
  g1[5] = (int)((unsigned)LDA);                              // tensor_dim0_stride lo32 (8B units)
  g1[6] = 0;
  g1[7] = 0;

  v4i z4 = {0, 0, 0, 0};
#if __clang_major__ >= 23
  v8i z8 = {0, 0, 0, 0, 0, 0, 0, 0};
  __builtin_amdgcn_tensor_load_to_lds(g0, g1, z4, z4, z8, 0);
#else
  __builtin_amdgcn_tensor_load_to_lds(g0, g1, z4, z4, 0);
#endif
#else
  (void)gsrc; (void)ldst;
#endif
}

// ---------------- init: zero fields, derive elastic constants ----------------
__global__ void cfrp_init_kernel(float* __restrict__ fields, unsigned long long nfloats,
                                 float* __restrict__ cc,
                                 const float* lc11, const float* lc22, const float* lc12,
                                 const float* lc16, const float* lc26, const float* lc66,
                                 const float* rho) {
  unsigned long long idx = (unsigned long long)blockIdx.x * blockDim.x + threadIdx.x;
  unsigned long long stride = (unsigned long long)gridDim.x * blockDim.x;
  for (unsigned long long k = idx; k < nfloats; k += stride) fields[k] = 0.0f;
  if (idx == 0) {
    float C11 = fminf(fmaxf(expf(lc11[0]), 1e9f), 1e13f);
    float C22 = fminf(fmaxf(expf(lc22[0]), 1e9f), 1e13f);
    float C12 = fminf(fmaxf(expf(lc12[0]), 1e9f), 1e13f);
    float C16 = fminf(fmaxf(expf(lc16[0]), 1e9f), 1e13f);
    float C26 = fminf(fmaxf(expf(lc26[0]), 1e9f), 1e13f);
    float C66 = fminf(fmaxf(expf(lc66[0]), 1e9f), 1e13f);
    cc[0] = C11;
    cc[1] = C22;
    cc[2] = C12 + C66;
    cc[3] = C16;
    cc[4] = 2.0f * C16;
    cc[5] = C26;
    cc[6] = 2.0f * C26;
    cc[7] = C66;
    cc[8] = 1.0f / rho[0];
  }
}

// ---------------- one leapfrog step (TDM-tiled stencil, rolling window) ----------------
__global__ __launch_bounds__(256)
void cfrp_step_kernel(const float2* __restrict__ cur,     // u(t-1), interleaved {ux,uy}
                      float2* __restrict__ dst,           // holds u(t-2); overwritten with u(t)
                      const float* __restrict__ cc,
                      const float* __restrict__ signal,
                      const float* __restrict__ gauss,
                      float* __restrict__ out, int t) {
  __shared__ float2 stile[HTILE * HTILE];

  const int gi0 = blockIdx.y * TILE;   // padded row of tile start (includes halo row)
  const int gj0 = blockIdx.x * TILE;   // padded col of tile start (includes halo col)

  // 256 threads: lane-contiguous tj (0..31), 8 row-groups of 4 consecutive rows.
  const int tj  = threadIdx.x & 31;
  const int tg  = threadIdx.x >> 5;    // 0..7
  const int ti0 = tg * 4;              // first tile-local output row of this thread
  const int j   = gj0 + tj;            // interior col 0..511

#ifdef HAVE_TDM
  if ((threadIdx.x >> 5) == 0) {       // wave 0 issues the single DMA for the block
    tdm_load_tile8(cur + (size_t)gi0 * LDA + gj0, stile);
  }
#else
  for (int k = threadIdx.x; k < HTILE * HTILE; k += 256) {
    int r = k / HTILE, c = k - r * HTILE;
    stile[k] = cur[(size_t)(gi0 + r) * LDA + (gj0 + c)];
  }
#endif

  // ---- overlap with the DMA: issue all LDS-independent global loads now ----
  size_t off[4];
  float2 prev[4];
  float  gv[4];
#pragma unroll
  for (int k = 0; k < 4; ++k) {
    const int i = gi0 + ti0 + k;                          // interior row 0..511
    off[k]  = (size_t)(i + 1) * LDA + (j + 1);            // padded float2 index
    prev[k] = dst[off[k]];                                // u(t-2), center only
    gv[k]   = gauss[i * NYg + j];
  }
  const float C11 = cc[0], C22 = cc[1], C12pC66 = cc[2], C16 = cc[3];
  const float twoC16 = cc[4], C26 = cc[5], twoC26 = cc[6], C66 = cc[7];
  const float inv_r = cc[8];
  const float famp = signal[t];

#ifdef HAVE_TDM
  if ((threadIdx.x >> 5) == 0) {
    __builtin_amdgcn_s_wait_tensorcnt(0);
  }
#endif
  __syncthreads();

  const float hinv2 = 1.0e6f;          // 1/H^2
  const float dt2 = 2.5e-15f;          // DT^2
  const bool emit = (t % STRIDEg) == 0;
  const int f = t / STRIDEg;

  // LDS row for interior tile row r is (r+1); window starts at the row above ti0.
  int lrow = ti0 * HTILE + (tj + 1);

  // rolling 3-row window of (W, C, E) float2 values
  float2 A_w = stile[lrow - 1], A_c = stile[lrow], A_e = stile[lrow + 1];   // north
  lrow += HTILE;
  float2 B_w = stile[lrow - 1], B_c = stile[lrow], B_e = stile[lrow + 1];   // center

#pragma unroll
  for (int k = 0; k < 4; ++k) {
    lrow += HTILE;
    const float2 C_w = stile[lrow - 1];
    const float2 C_c = stile[lrow];
    const float2 C_e = stile[lrow + 1];                                     // south

    // x = row (i) direction, y = col (j) direction, matching the reference
    const float dxx_x = (C_c.x + A_c.x - 2.0f * B_c.x) * hinv2;
    const float dyy_x = (B_e.x + B_w.x - 2.0f * B_c.x) * hinv2;
    const float dxy_x = 0.25f * hinv2 * (A_w.x - A_e.x - C_w.x + C_e.x);
    const float dxx_y = (C_c.y + A_c.y - 2.0f * B_c.y) * hinv2;
    const float dyy_y = (B_e.y + B_w.y - 2.0f * B_c.y) * hinv2;
    const float dxy_y = 0.25f * hinv2 * (A_w.y - A_e.y - C_w.y + C_e.y);

    const float Lux = (C11 * dxx_x + C66 * dyy_x + C12pC66 * dxy_y +
                       twoC16 * dxy_x + C16 * dxx_y + C26 * dyy_y) * inv_r;
    const float fy = famp * gv[k];
    const float Luy = (C66 * dxx_y + C22 * dyy_y + C12pC66 * dxy_x +
                       C16 * dxx_x + C26 * dyy_x + twoC26 * dxy_y + fy) * inv_r;

    float2 nv;
    nv.x = 2.0f * B_c.x - prev[k].x + dt2 * Lux;
    nv.y = 2.0f * B_c.y - prev[k].y + dt2 * Luy;
    dst[off[k]] = nv;                                     // in-place leapfrog rotate

    if (emit) {
      const int i = gi0 + ti0 + k;
      const size_t oidx = (size_t)f * FRAME_ELEMS + (size_t)i * NYg + j;
      out[oidx] = nv.x;
      out[(size_t)FRAMES * FRAME_ELEMS + oidx] = nv.y;
    }

    // slide window south
    A_w = B_w; A_c = B_c; A_e = B_e;
    B_w = C_w; B_c = C_c; B_e = C_e;
  }
}

// ---------------- launch ----------------
extern "C" void kernel_launch(void* const* d_in, const int* in_sizes, int n_in,
                              void* d_out, int out_size, void* d_ws, size_t ws_size,
                              hipStream_t stream) {
  (void)in_sizes; (void)n_in; (void)out_size; (void)ws_size;

  const float* lc11 = (const float*)d_in[0];
  const float* lc22 = (const float*)d_in[1];
  const float* lc12 = (const float*)d_in[2];
  const float* lc16 = (const float*)d_in[3];
  const float* lc26 = (const float*)d_in[4];
  const float* lc66 = (const float*)d_in[5];
  const float* rho  = (const float*)d_in[6];
  const float* signal = (const float*)d_in[7];
  const float* gauss  = (const float*)d_in[8];
  float* out = (float*)d_out;

  char* ws = (char*)d_ws;
  float* cc = (float*)ws;                       // 16 floats of derived constants
  float2* fields = (float2*)(ws + 64);
  const size_t BUF = (size_t)LDA * PADH;        // float2 elements per padded field buffer
  float2* b0 = fields;
  float2* b1 = fields + BUF;

  cfrp_init_kernel<<<512, 256, 0, stream>>>((float*)fields,
                                            (unsigned long long)(4 * BUF),  // 2 buffers * 2 floats
                                            cc, lc11, lc22, lc12, lc16, lc26, lc66, rho);

  dim3 grid(NYg / TILE, NXg / TILE);            // x: cols, y: rows
  for (int t = 0; t < NTg; ++t) {
    // invariant: u(t-1) in buf[t&1], u(t-2) in buf[1-(t&1)]; write u(t) over u(t-2)
    float2* cur = (t & 1) ? b1 : b0;
    float2* dstp = (t & 1) ? b0 : b1;
    cfrp_step_kernel<<<grid, 256, 0, stream>>>(cur, dstp, cc, signal, gauss, out, t);
  }
}